// ConvDipModel_7997229105888
// MI455X (gfx1250) — compile-verified
//
#include <hip/hip_runtime.h>
#include <math.h>

// ---------------------------------------------------------------------------
// CDNA5 (gfx1250) ConvDip pipeline. GEMMs on matrix cores via
// V_WMMA_F32_16X16X4_F32, K-permuted contiguous float4 loads, and
// unroll-by-2 double-buffered K-chunks (16 k-values / 4 wmma steps each).
// ---------------------------------------------------------------------------

typedef float v2f __attribute__((ext_vector_type(2)));
typedef float v8f __attribute__((ext_vector_type(8)));

__device__ __forceinline__ v2f f4lo(const float4& f) { v2f r; r.x = f.x; r.y = f.y; return r; }
__device__ __forceinline__ v2f f4hi(const float4& f) { v2f r; r.x = f.z; r.y = f.w; return r; }

template <int NW>
struct Bufs {
    float4 a0, a1;
    float4 b0[NW], b1[NW];
};

// C[M,N] = A[M,K] @ W[N,K]^T, epilogue: MODE==0 -> C *= epi[n]; MODE==1 -> C += epi[n]
// One wave computes a 16 x (16*NW) strip. K % 32 == 0, M % 16 == 0.
template <int MODE, int NW>
__global__ __launch_bounds__(256) void gemm_wmma_k(
    const float* __restrict__ A, const float* __restrict__ W,
    const float* __restrict__ epi, float* __restrict__ C,
    int M, int N, int K, int nStrips)
{
    const int lane  = threadIdx.x & 31;
    const int wave  = blockIdx.x * (blockDim.x >> 5) + (threadIdx.x >> 5);
    const int mTile = wave / nStrips;
    const int strip = wave - mTile * nStrips;
    if (mTile * 16 >= M) return;               // wave-uniform exit (EXEC stays full)

    const int half  = lane >> 4;               // 0: lanes 0-15, 1: lanes 16-31
    const int lidx  = lane & 15;
    // K-permutation: within each 16-k chunk, lanes 0-15 consume k0+0..7,
    // lanes 16-31 consume k0+8..15 (contiguous per lane -> b128 loads).
    const int koff  = half * 8;
    const int nBase = strip * (NW * 16);

    const float* Ap = A + (size_t)(mTile * 16 + lidx) * K + koff;
    const float* Wp[NW];
#pragma unroll
    for (int t = 0; t < NW; ++t) {
        int n  = nBase + t * 16 + lidx;
        int nc = (n < N) ? n : (N - 1);        // clamp: safe load, never stored
        Wp[t]  = W + (size_t)nc * K + koff;
    }

    v8f acc[NW] = {};

    auto loadChunk = [&](Bufs<NW>& f, int k) {
        f.a0 = *(const float4*)(Ap + k);
        f.a1 = *(const float4*)(Ap + k + 4);
#pragma unroll
        for (int t = 0; t < NW; ++t) {
            f.b0[t] = *(const float4*)(Wp[t] + k);
            f.b1[t] = *(const float4*)(Wp[t] + k + 4);
        }
    };
    auto computeChunk = [&](const Bufs<NW>& f) {
        v2f af[4];
        af[0] = f4lo(f.a0); af[1] = f4hi(f.a0);
        af[2] = f4lo(f.a1); af[3] = f4hi(f.a1);
#pragma unroll
        for (int s = 0; s < 4; ++s) {
#pragma unroll
            for (int t = 0; t < NW; ++t) {   // t-inner: rotate accumulators
                v2f b = (s == 0) ? f4lo(f.b0[t]) :
                        (s == 1) ? f4hi(f.b0[t]) :
                        (s == 2) ? f4lo(f.b1[t]) : f4hi(f.b1[t]);
                // 8 args: (neg_a, A, neg_b, B, c_mod, C, reuse_a, reuse_b)
                acc[t] = __builtin_amdgcn_wmma_f32_16x16x4_f32(
                    false, af[s], false, b, (short)0, acc[t], false, false);
            }
        }
    };

    // ---- unroll-by-2 double-buffered K loop (K % 32 == 0) ----
    Bufs<NW> buf0, buf1;
    loadChunk(buf0, 0);
    for (int k = 0; k < K; k += 32) {
        loadChunk(buf1, k + 16);               // always in range (K % 32 == 0)
        computeChunk(buf0);
        if (k + 32 < K) loadChunk(buf0, k + 32);
        computeChunk(buf1);
    }

    // C/D layout: VGPR r -> (M = half*8 + r, N = lidx)
    const int mOut = mTile * 16 + half * 8;
#pragma unroll
    for (int t = 0; t < NW; ++t) {
        int n = nBase + t * 16 + lidx;
        if (n >= N) continue;
        float e = epi[n];
#pragma unroll
        for (int r = 0; r < 8; ++r) {
            float v = acc[t][r];
            v = (MODE == 0) ? v * e : v + e;
            C[(size_t)(mOut + r) * N + n] = v;
        }
    }
}

// 3x3 SAME conv on 12x12 grid, 1 -> 8 channels, + bias.
// Y layout: [b][c*144 + i*12 + j]  (matches torch .view flatten order)
__global__ __launch_bounds__(256) void conv_k(
    const float* __restrict__ G, const float* __restrict__ w,
    const float* __restrict__ bias, float* __restrict__ Y, int total)
{
    __shared__ float sw[72];
    __shared__ float sb[8];
    if (threadIdx.x < 72) sw[threadIdx.x] = w[threadIdx.x];
    if (threadIdx.x < 8)  sb[threadIdx.x] = bias[threadIdx.x];
    __syncthreads();

    int idx = blockIdx.x * 256 + threadIdx.x;
    if (idx >= total) return;
    int b = idx / 1152, r = idx - b * 1152;
    int c = r / 144,    p = r - c * 144;
    int i = p / 12,     j = p - i * 12;
    const float* g = G + (size_t)b * 144;
    float s = sb[c];
#pragma unroll
    for (int di = 0; di < 3; ++di) {
        int ii = i + di - 1;
        if (ii < 0 || ii >= 12) continue;
#pragma unroll
        for (int dj = 0; dj < 3; ++dj) {
            int jj = j + dj - 1;
            if (jj < 0 || jj >= 12) continue;
            s += g[ii * 12 + jj] * sw[c * 9 + di * 3 + dj];
        }
    }
    Y[idx] = s;
}

// ---- BN1 (per-channel over B*144): deterministic two-pass, double accum ----
__global__ __launch_bounds__(256) void bn1_part_k(
    const float* __restrict__ Y, double* __restrict__ part)
{
    int chunk = blockIdx.x, c = blockIdx.y, t = threadIdx.x;
    double s = 0.0, q = 0.0;
    if (t < 144) {
        for (int r = 0; r < 256; ++r) {
            float v = Y[(size_t)(chunk * 256 + r) * 1152 + c * 144 + t];
            s += v; q += (double)v * v;
        }
    }
    __shared__ double ls[256], lq[256];
    ls[t] = s; lq[t] = q; __syncthreads();
    for (int off = 128; off > 0; off >>= 1) {
        if (t < off) { ls[t] += ls[t + off]; lq[t] += lq[t + off]; }
        __syncthreads();
    }
    if (t == 0) {
        part[(c * gridDim.x + chunk) * 2 + 0] = ls[0];
        part[(c * gridDim.x + chunk) * 2 + 1] = lq[0];
    }
}

__global__ void bn1_fin_k(const double* __restrict__ part,
                          const float* __restrict__ g, const float* __restrict__ b,
                          float* __restrict__ a1, float* __restrict__ b1,
                          int chunks, double nElem)
{
    int c = threadIdx.x;
    if (c >= 8) return;
    double s = 0.0, q = 0.0;
    for (int k = 0; k < chunks; ++k) {
        s += part[(c * chunks + k) * 2 + 0];
        q += part[(c * chunks + k) * 2 + 1];
    }
    double mu = s / nElem;
    double var = q / nElem - mu * mu;
    float aa = (float)((double)g[c] / sqrt(var + 1e-5));
    a1[c] = aa;
    b1[c] = b[c] - (float)mu * aa;
}

__global__ __launch_bounds__(256) void bn1_apply_k(
    float* __restrict__ Y, const float* __restrict__ a1,
    const float* __restrict__ b1, int total)
{
    int idx = blockIdx.x * 256 + threadIdx.x;
    if (idx >= total) return;
    int ch = (idx % 1152) / 144;
    float v = a1[ch] * Y[idx] + b1[ch];
    Y[idx] = v > 0.0f ? v : 0.0f;
}

// ---- BN2 (per-feature over B, 512 features) ----
__global__ __launch_bounds__(256) void bn2_part_k(
    const float* __restrict__ H, double* __restrict__ part)
{
    int chunk = blockIdx.x, t = threadIdx.x;
    double s0 = 0, q0 = 0, s1 = 0, q1 = 0;
    for (int r = 0; r < 256; ++r) {
        size_t base = (size_t)(chunk * 256 + r) * 512;
        float v0 = H[base + t];
        float v1 = H[base + t + 256];
        s0 += v0; q0 += (double)v0 * v0;
        s1 += v1; q1 += (double)v1 * v1;
    }
    part[((size_t)chunk * 512 + t) * 2 + 0]       = s0;
    part[((size_t)chunk * 512 + t) * 2 + 1]       = q0;
    part[((size_t)chunk * 512 + t + 256) * 2 + 0] = s1;
    part[((size_t)chunk * 512 + t + 256) * 2 + 1] = q1;
}

__global__ __launch_bounds__(256) void bn2_fin_k(
    const double* __restrict__ part, const float* __restrict__ g,
    const float* __restrict__ b, float* __restrict__ a2, float* __restrict__ b2,
    int chunks, double nElem)
{
    int t = threadIdx.x;
    for (int f = t; f < 512; f += 256) {
        double s = 0.0, q = 0.0;
        for (int k = 0; k < chunks; ++k) {
            s += part[((size_t)k * 512 + f) * 2 + 0];
            q += part[((size_t)k * 512 + f) * 2 + 1];
        }
        double mu = s / nElem;
        double var = q / nElem - mu * mu;
        float aa = (float)((double)g[f] / sqrt(var + 1e-5));
        a2[f] = aa;
        b2[f] = b[f] - (float)mu * aa;
    }
}

__global__ __launch_bounds__(256) void bn2_apply_k(
    float* __restrict__ H, const float* __restrict__ a2,
    const float* __restrict__ b2, int total)
{
    int idx = blockIdx.x * 256 + threadIdx.x;
    if (idx >= total) return;
    int f = idx & 511;
    float v = a2[f] * H[idx] + b2[f];
    H[idx] = v > 0.0f ? v : 0.0f;
}

// ---------------------------------------------------------------------------
extern "C" void kernel_launch(void* const* d_in, const int* in_sizes, int n_in,
                              void* d_out, int out_size, void* d_ws, size_t ws_size,
                              hipStream_t stream)
{
    const float* x        = (const float*)d_in[0];
    const float* interp_W = (const float*)d_in[1];
    const float* head_mask= (const float*)d_in[2];
    const float* conv_w   = (const float*)d_in[3];
    const float* conv_b   = (const float*)d_in[4];
    const float* bn1_g    = (const float*)d_in[5];
    const float* bn1_b    = (const float*)d_in[6];
    const float* fc1_w    = (const float*)d_in[7];
    const float* fc1_b    = (const float*)d_in[8];
    const float* bn2_g    = (const float*)d_in[9];
    const float* bn2_b    = (const float*)d_in[10];
    const float* fc2_w    = (const float*)d_in[11];
    const float* fc2_b    = (const float*)d_in[12];
    float* out = (float*)d_out;

    const int B      = in_sizes[0] / 64;   // 16384
    const int N1     = 144, K1 = 64;
    const int N2     = 512, K2 = 1152;
    const int N3     = 5124, K3 = 512;
    const int chunks = B / 256;            // 64

    // workspace layout (needs ~119 MB)
    char*  ws   = (char*)d_ws;
    size_t off  = 0;
    auto alloc = [&](size_t bytes) { void* p = ws + off; off = (off + bytes + 255) & ~(size_t)255; return p; };
    float*  G     = (float*) alloc((size_t)B * 144 * 4);
    float*  Y     = (float*) alloc((size_t)B * 1152 * 4);
    float*  H     = (float*) alloc((size_t)B * 512 * 4);
    double* part1 = (double*)alloc((size_t)8 * chunks * 2 * 8);
    double* part2 = (double*)alloc((size_t)chunks * 512 * 2 * 8);
    float*  a1    = (float*) alloc(8 * 4);
    float*  b1    = (float*) alloc(8 * 4);
    float*  a2    = (float*) alloc(512 * 4);
    float*  b2    = (float*) alloc(512 * 4);
    (void)ws_size; (void)n_in; (void)out_size;

    const int mTiles = B / 16;  // 1024

    // 1) G = (x @ interp_W^T) * head_mask     (N=144 -> 9 tiles, NW=3 -> 3 strips)
    {
        int nStrips = 3;
        int waves = mTiles * nStrips;
        gemm_wmma_k<0, 3><<<(waves * 32 + 255) / 256, 256, 0, stream>>>(
            x, interp_W, head_mask, G, B, N1, K1, nStrips);
    }
    // 2) conv 3x3 + bias -> Y
    {
        int total = B * 1152;
        conv_k<<<(total + 255) / 256, 256, 0, stream>>>(G, conv_w, conv_b, Y, total);
    }
    // 3-5) BN1 stats + apply (relu fused)
    bn1_part_k<<<dim3(chunks, 8), 256, 0, stream>>>(Y, part1);
    bn1_fin_k<<<1, 32, 0, stream>>>(part1, bn1_g, bn1_b, a1, b1, chunks, (double)B * 144.0);
    bn1_apply_k<<<(B * 1152 + 255) / 256, 256, 0, stream>>>(Y, a1, b1, B * 1152);
    // 6) H = Y' @ fc1_w^T + fc1_b             (N=512 -> 32 tiles, NW=4 -> 8 strips)
    {
        int nStrips = 8;
        int waves = mTiles * nStrips;
        gemm_wmma_k<1, 4><<<(waves * 32 + 255) / 256, 256, 0, stream>>>(
            Y, fc1_w, fc1_b, H, B, N2, K2, nStrips);
    }
    // 7-9) BN2 stats + apply (relu fused)
    bn2_part_k<<<chunks, 256, 0, stream>>>(H, part2);
    bn2_fin_k<<<1, 256, 0, stream>>>(part2, bn2_g, bn2_b, a2, b2, chunks, (double)B);
    bn2_apply_k<<<(B * 512 + 255) / 256, 256, 0, stream>>>(H, a2, b2, B * 512);
    // 10) out = H' @ fc2_w^T + fc2_b          (N=5124 -> 321 tiles, NW=4 -> 81 strips)
    {
        int nTiles  = (N3 + 15) / 16;            // 321
        int nStrips = (nTiles + 3) / 4;          // 81
        int waves = mTiles * nStrips;
        gemm_wmma_k<1, 4><<<(waves * 32 + 255) / 256, 256, 0, stream>>>(
            H, fc2_w, fc2_b, out, B, N3, K3, nStrips);
    }
}